// RoiAlign_75333726372203
// MI455X (gfx1250) — compile-verified
//
#include <hip/hip_runtime.h>

// Problem constants (match reference)
#define R_   512
#define C_   128
#define H_   64
#define W_   64
#define T_   64
#define PH_  7
#define PW_  7
#define PT_  7
#define IMG_ 64.0f

// K ordering: k = pt*8 + j*2 + h   (pt: 0..6 z-point, j: (y,x) corner, h: z lo/hi)
#define KTOT    56
#define GSTRIDE 137   // LDS row stride in floats; 137 mod 64 = 9 (coprime) -> conflict-free

typedef float v2f __attribute__((ext_vector_type(2)));
typedef float v8f __attribute__((ext_vector_type(8)));

__global__ __launch_bounds__(256)
void roi_align3d_wmma(const float* __restrict__ feat,
                      const float* __restrict__ rois,
                      float* __restrict__ out)
{
    const int bx  = blockIdx.x;
    const int r   = bx / (PH_ * PW_);
    const int rm  = bx - r * (PH_ * PW_);
    const int ph  = rm / PW_;
    const int pw  = rm - ph * PW_;
    const int tid = threadIdx.x;

    __shared__ float sG[KTOT * GSTRIDE];          // gathered corners: [k][c]
    __shared__ int   sYI[2], sXI[2], sZI[14], sB;
    __shared__ float sWY[2], sWX[2], sWZ[14];

    const float* roi = rois + r * 7;

    // ---- per-block interpolation setup ----
    if (tid < 7) {                                // z grid: pt = tid
        const int   pt = tid;
        const float lo = roi[2] * (1.0f / IMG_);
        const float hi = roi[5] * (1.0f / IMG_);
        const float c  = lo * (T_ - 1) + ((float)pt / (PT_ - 1)) * (hi - lo) * (T_ - 1);
        const float c0 = floorf(c);
        const float w  = c - c0;
        sZI[2 * pt]     = min(max((int)c0,     0), T_ - 1);
        sZI[2 * pt + 1] = min(max((int)c0 + 1, 0), T_ - 1);
        sWZ[2 * pt]     = 1.0f - w;
        sWZ[2 * pt + 1] = w;
    } else if (tid == 7) {                        // y grid for this ph
        const float lo = roi[0] * (1.0f / IMG_);
        const float hi = roi[3] * (1.0f / IMG_);
        const float c  = lo * (H_ - 1) + ((float)ph / (PH_ - 1)) * (hi - lo) * (H_ - 1);
        const float c0 = floorf(c);
        const float w  = c - c0;
        sYI[0] = min(max((int)c0,     0), H_ - 1);
        sYI[1] = min(max((int)c0 + 1, 0), H_ - 1);
        sWY[0] = 1.0f - w;
        sWY[1] = w;
    } else if (tid == 8) {                        // x grid for this pw
        const float lo = roi[1] * (1.0f / IMG_);
        const float hi = roi[4] * (1.0f / IMG_);
        const float c  = lo * (W_ - 1) + ((float)pw / (PW_ - 1)) * (hi - lo) * (W_ - 1);
        const float c0 = floorf(c);
        const float w  = c - c0;
        sXI[0] = min(max((int)c0,     0), W_ - 1);
        sXI[1] = min(max((int)c0 + 1, 0), W_ - 1);
        sWX[0] = 1.0f - w;
        sWX[1] = w;
    } else if (tid == 9) {
        sB = (int)roi[6];
    }
    __syncthreads();

    // ---- cooperative gather of the 56x128 corner matrix into LDS ----
    // k = pt*8 + j*2 + h : decode -> y = sYI[(k>>2)&1], x = sXI[(k>>1)&1], z = sZI[2*(k>>3)+(k&1)]
    const float* fb = feat + (long)sB * ((long)C_ * H_ * W_ * T_);
    #pragma unroll 4
    for (int i = 0; i < (KTOT * C_) / 256; ++i) {
        const int v  = i * 256 + tid;
        const int c  = v / KTOT;
        const int k  = v - c * KTOT;
        const long off = (long)c * (H_ * W_ * T_)
                       + sYI[(k >> 2) & 1] * (W_ * T_)
                       + sXI[(k >> 1) & 1] * T_
                       + sZI[2 * (k >> 3) + (k & 1)];
        sG[k * GSTRIDE + c] = fb[off];
    }
    __syncthreads();

    const int  lane   = tid & 31;
    const int  wv     = tid >> 5;                 // 0..7 -> 16-channel tile
    const int  col    = lane & 15;                // N (channel) == M (row id) for this lane
    const int  cT     = wv * 16;
    const bool lohalf = (lane < 16);              // lanes 0-15 hold K0/K1, 16-31 hold K2/K3
    const float* gb   = sG + (lohalf ? 0 : 2) * GSTRIDE + cT + col;

    // ---- per-lane weight table: the only 8 weights this lane's row can use ----
    // wm[j*2+h] = wy[jy]*wx[jx] * wz(row, h); valid for row==col<7, masked otherwise.
    const int  cc    = min(col, 6);
    const float wzlo = sWZ[2 * cc];
    const float wzhi = sWZ[2 * cc + 1];
    float wYX[4];
    wYX[0] = sWY[0] * sWX[0];
    wYX[1] = sWY[0] * sWX[1];
    wYX[2] = sWY[1] * sWX[0];
    wYX[3] = sWY[1] * sWX[1];
    float wm[8];
    #pragma unroll
    for (int t = 0; t < 8; ++t) wm[t] = wYX[t >> 1] * ((t & 1) ? wzhi : wzlo);

    // ---- WMMA: out[pt, c] = sum_k W[pt,k] * G[k,c]  (exact f32) ----
    const int q0 = lohalf ? 0 : 2;                // this lane-half's K slots within a chunk

    v8f acc = {};
    #pragma unroll
    for (int kb = 0; kb < KTOT / 4; ++kb) {
        const int  pt  = kb >> 1;                 // chunk's pt (compile-time)
        const int  tb  = 4 * (kb & 1);            // chunk's (j,h) base (compile-time)
        const bool hit = (col == pt);             // hoisted v_cmp per pt, reused

        v2f a;                                    // one cndmask pair per chunk
        a.x = hit ? wm[tb + q0]     : 0.0f;
        a.y = hit ? wm[tb + q0 + 1] : 0.0f;

        v2f b;                                    // gathered features, const offsets from gb
        b.x = gb[(4 * kb) * GSTRIDE];
        b.y = gb[(4 * kb + 1) * GSTRIDE];

        acc = __builtin_amdgcn_wmma_f32_16x16x4_f32(false, a, false, b,
                                                    (short)0, acc, false, false);
    }

    // ---- epilogue: lanes 0-15 hold D rows M=0..7; rows 0..6 are the 7 pt outputs ----
    if (lane < 16) {
        const int c = cT + col;
        float* o = out + ((long)(r * C_ + c)) * (PH_ * PW_ * PT_)
                       + ph * (PW_ * PT_) + pw * PT_;
        #pragma unroll
        for (int pt = 0; pt < PT_; ++pt) o[pt] = acc[pt];
    }
}

extern "C" void kernel_launch(void* const* d_in, const int* in_sizes, int n_in,
                              void* d_out, int out_size, void* d_ws, size_t ws_size,
                              hipStream_t stream) {
    const float* feat = (const float*)d_in[0];   // (B,C,H,W,T) f32
    const float* rois = (const float*)d_in[1];   // (R,7) f32
    float* out = (float*)d_out;                  // (R,C,PH,PW,PT) f32

    dim3 grid(R_ * PH_ * PW_);                   // one block per (roi, ph, pw)
    roi_align3d_wmma<<<grid, 256, 0, stream>>>(feat, rois, out);
    (void)in_sizes; (void)n_in; (void)out_size; (void)d_ws; (void)ws_size;
}